// Local_Encoder_67302137528732
// MI455X (gfx1250) — compile-verified
//
#include <hip/hip_runtime.h>

typedef __attribute__((ext_vector_type(2))) float v2f;
typedef __attribute__((ext_vector_type(8))) float v8f;

#define KNN 16
#define LRELU_SLOPE 0.2f
#define GN_EPS 1e-5f

// ---------------------------------------------------------------------------
// fea = w_in(16x3) @ coor(3xN) + b_in
// ---------------------------------------------------------------------------
__global__ void fea_in_kernel(const float* __restrict__ coor,
                              const float* __restrict__ w_in,
                              const float* __restrict__ b_in,
                              float* __restrict__ fea, int N) {
  int i = blockIdx.x * blockDim.x + threadIdx.x;
  if (i >= 16 * N) return;
  int o = i / N, n = i - o * N;
  float acc = b_in[o];
  acc += w_in[o * 3 + 0] * coor[n];
  acc += w_in[o * 3 + 1] * coor[N + n];
  acc += w_in[o * 3 + 2] * coor[2 * N + n];
  fea[i] = acc;
}

// ---------------------------------------------------------------------------
// Exact kNN (K=16): one thread per query, keys staged through LDS in tiles,
// register-resident sorted insertion list (ascending distance, ties -> lower
// index first, matching lax.top_k ordering on -d2).
// ---------------------------------------------------------------------------
__global__ void knn_kernel(const float* __restrict__ q, int qpitch, int Nq,
                           const float* __restrict__ k, int kpitch, int Nk,
                           int* __restrict__ idx_out) {
  __shared__ float kx[256], ky[256], kz[256];
  int qi = blockIdx.x * blockDim.x + threadIdx.x;
  float qx = 0.f, qy = 0.f, qz = 0.f;
  if (qi < Nq) {
    qx = q[qi];
    qy = q[qpitch + qi];
    qz = q[2 * qpitch + qi];
  }
  float bd[KNN];
  int bi[KNN];
#pragma unroll
  for (int j = 0; j < KNN; ++j) { bd[j] = 3.0e38f; bi[j] = 0; }

  for (int base = 0; base < Nk; base += 256) {
    __syncthreads();
    int t = base + threadIdx.x;
    if (t < Nk) {
      kx[threadIdx.x] = k[t];
      ky[threadIdx.x] = k[kpitch + t];
      kz[threadIdx.x] = k[2 * kpitch + t];
    }
    __syncthreads();
    int lim = (Nk - base < 256) ? (Nk - base) : 256;
    if (qi < Nq) {
      for (int j = 0; j < lim; ++j) {
        float dx = qx - kx[j], dy = qy - ky[j], dz = qz - kz[j];
        float d = dx * dx + dy * dy + dz * dz;
        if (d < bd[KNN - 1]) {
          int p = KNN - 1;
          while (p > 0 && bd[p - 1] > d) {
            bd[p] = bd[p - 1];
            bi[p] = bi[p - 1];
            --p;
          }
          bd[p] = d;
          bi[p] = base + j;
        }
      }
    }
  }
  if (qi < Nq) {
#pragma unroll
    for (int j = 0; j < KNN; ++j) idx_out[j * Nq + qi] = bi[j];
  }
}

// ---------------------------------------------------------------------------
// Farthest point sampling: one block; each of 256 threads owns <=16 points
// cached in registers; LDS argmax tree with first-index tie-break (argmax).
// ---------------------------------------------------------------------------
__global__ void fps_kernel(const float* __restrict__ p, int pitch, int N,
                           int m, int* __restrict__ out) {
  const int T = 256;
  int tid = threadIdx.x;
  int npt = (N + T - 1) / T;  // <= 16
  float px[16], py[16], pz[16], d[16];
#pragma unroll
  for (int j = 0; j < 16; ++j) { px[j] = py[j] = pz[j] = 0.f; d[j] = 1e10f; }
  for (int j = 0; j < npt; ++j) {
    int n = j * T + tid;
    if (n < N) { px[j] = p[n]; py[j] = p[pitch + n]; pz[j] = p[2 * pitch + n]; }
  }
  __shared__ float sx, sy, sz;
  __shared__ float sv[256];
  __shared__ int si[256];
  if (tid == 0) { sx = p[0]; sy = p[pitch]; sz = p[2 * pitch]; out[0] = 0; }
  __syncthreads();
  for (int i = 1; i < m; ++i) {
    float lx = sx, ly = sy, lz = sz;
    float best = -1.f;
    int bidx = 0x7fffffff;
    for (int j = 0; j < npt; ++j) {
      int n = j * T + tid;
      if (n < N) {
        float dx = px[j] - lx, dy = py[j] - ly, dz = pz[j] - lz;
        float dd = dx * dx + dy * dy + dz * dz;
        if (dd < d[j]) d[j] = dd;
        if (d[j] > best) { best = d[j]; bidx = n; }
      }
    }
    sv[tid] = best;
    si[tid] = bidx;
    __syncthreads();
    for (int off = 128; off > 0; off >>= 1) {
      if (tid < off) {
        if (sv[tid + off] > sv[tid] ||
            (sv[tid + off] == sv[tid] && si[tid + off] < si[tid])) {
          sv[tid] = sv[tid + off];
          si[tid] = si[tid + off];
        }
      }
      __syncthreads();
    }
    if (tid == 0) {
      int w = si[0];
      out[i] = w;
      sx = p[w];
      sy = p[pitch + w];
      sz = p[2 * pitch + w];
    }
    __syncthreads();
  }
}

// coorQ(3 x m, pitch m) = cur[:, fi]
__global__ void gather_coor_kernel(const float* __restrict__ cur, int pitch,
                                   const int* __restrict__ fi,
                                   float* __restrict__ cq, int m) {
  int j = blockIdx.x * blockDim.x + threadIdx.x;
  if (j >= m) return;
  int n = fi[j];
  cq[j] = cur[n];
  cq[m + j] = cur[pitch + n];
  cq[2 * m + j] = cur[2 * pitch + n];
}

// ---------------------------------------------------------------------------
// graph feature: X[CinPad x (Nq*16)]; rows 0..2 = relative coords, rows
// 3..3+Cf-1 = gathered neighbor features, rows >= 3+Cf zero-padded so the
// WMMA GEMM can run an unguarded K loop on CinPad (multiple of 4).
// ---------------------------------------------------------------------------
__global__ void graph_feat_kernel(const float* __restrict__ cq, int qpitch, int Nq,
                                  const float* __restrict__ ck, int kpitch,
                                  const float* __restrict__ fea, int fpitch, int Cf,
                                  int CinPad, const int* __restrict__ idx,
                                  float* __restrict__ X) {
  int M = Nq * 16;
  int col = blockIdx.x * blockDim.x + threadIdx.x;
  if (col >= M) return;
  int qi = col >> 4, kk = col & 15;
  int nb = idx[kk * Nq + qi];
  X[col] = ck[nb] - cq[qi];
  X[(size_t)M + col] = ck[kpitch + nb] - cq[qpitch + qi];
  X[2 * (size_t)M + col] = ck[2 * kpitch + nb] - cq[2 * qpitch + qi];
  for (int c = 0; c < Cf; ++c)
    X[(size_t)(3 + c) * M + col] = fea[(size_t)c * fpitch + nb];
  for (int c = 3 + Cf; c < CinPad; ++c)
    X[(size_t)c * M + col] = 0.f;
}

// W(Cout x Cin) -> Wp(Cout x CinPad), zero pad
__global__ void pad_w_kernel(const float* __restrict__ W,
                             float* __restrict__ Wp,
                             int Cout, int Cin, int CinPad) {
  int i = blockIdx.x * blockDim.x + threadIdx.x;
  if (i >= Cout * CinPad) return;
  int r = i / CinPad, c = i - r * CinPad;
  Wp[i] = (c < Cin) ? W[r * Cin + c] : 0.f;
}

// ---------------------------------------------------------------------------
// 1x1 conv as GEMM on the FP32 matrix core: Y[Cout,M] = W[Cout,Cin4] @ X[Cin4,M]
// V_WMMA_F32_16X16X4_F32. Cin4 is a multiple of 4 and operands are zero-padded,
// so the K loop is branch-free. Each wave computes a 16x64 strip: the A (weight)
// fragment is loaded once per K-step (64-bit load) and reused by 4 WMMAs.
// A layout (16x4 f32): lanes 0-15 row M, v0=K0 v1=K1; lanes 16-31 K2/K3.
// B layout (4x16 f32): lanes 0-15 col N, v0=K0 v1=K1; lanes 16-31 K2/K3.
// D layout: vgpr r -> row r (lanes 0-15) / row r+8 (lanes 16-31), col = lane&15.
// ---------------------------------------------------------------------------
__global__ void conv1x1_wmma(const float* __restrict__ W,
                             const float* __restrict__ X,
                             float* __restrict__ Y,
                             int Cout, int Cin4, int M) {
  int lane = threadIdx.x & 31;
  int wave = threadIdx.x >> 5;
  int col0 = (blockIdx.x * 4 + wave) * 64 + (lane & 15);
  int rowA = blockIdx.y * 16 + (lane & 15);
  int hi2 = (lane >= 16) ? 2 : 0;
  const float* wrow = W + (size_t)rowA * Cin4 + hi2;

  v8f acc0 = {}, acc1 = {}, acc2 = {}, acc3 = {};
  for (int k0 = 0; k0 < Cin4; k0 += 4) {
    v2f a = *(const v2f*)(wrow + k0);               // A[row][k0+hi2 .. +1]
    const float* xr0 = X + (size_t)(k0 + hi2) * M + col0;
    const float* xr1 = xr0 + M;
    v2f b0, b1, b2, b3;
    b0.x = xr0[0];  b0.y = xr1[0];
    b1.x = xr0[16]; b1.y = xr1[16];
    b2.x = xr0[32]; b2.y = xr1[32];
    b3.x = xr0[48]; b3.y = xr1[48];
    acc0 = __builtin_amdgcn_wmma_f32_16x16x4_f32(false, a, false, b0,
                                                 (short)0, acc0, false, false);
    acc1 = __builtin_amdgcn_wmma_f32_16x16x4_f32(false, a, false, b1,
                                                 (short)0, acc1, false, false);
    acc2 = __builtin_amdgcn_wmma_f32_16x16x4_f32(false, a, false, b2,
                                                 (short)0, acc2, false, false);
    acc3 = __builtin_amdgcn_wmma_f32_16x16x4_f32(false, a, false, b3,
                                                 (short)0, acc3, false, false);
  }
  int rbase = blockIdx.y * 16 + ((lane >= 16) ? 8 : 0);
#pragma unroll
  for (int r = 0; r < 8; ++r) {
    size_t o = (size_t)(rbase + r) * M + col0;
    Y[o]      = acc0[r];
    Y[o + 16] = acc1[r];
    Y[o + 32] = acc2[r];
    Y[o + 48] = acc3[r];
  }
}

// ---------------------------------------------------------------------------
// GroupNorm: deterministic two-stage reduction. Stage 1: 4 groups x 64 chunks
// of partial (sum, sumsq). Stage 2: tree-reduce the 64 partials per group.
// ---------------------------------------------------------------------------
__global__ void gn_partial_kernel(const float* __restrict__ Y,
                                  float* __restrict__ part, int Cout, int M) {
  int g = blockIdx.x;       // group 0..3
  int chunk = blockIdx.y;   // 0..63
  int cg = Cout / 4;
  size_t total = (size_t)cg * M;
  const float* base = Y + (size_t)g * cg * M;
  size_t beg = total * chunk / 64;
  size_t end = total * (chunk + 1) / 64;
  float s = 0.f, s2 = 0.f;
  for (size_t i = beg + threadIdx.x; i < end; i += blockDim.x) {
    float v = base[i];
    s += v;
    s2 += v * v;
  }
  __shared__ float sh[256], sh2[256];
  sh[threadIdx.x] = s;
  sh2[threadIdx.x] = s2;
  __syncthreads();
  for (int off = 128; off > 0; off >>= 1) {
    if ((int)threadIdx.x < off) {
      sh[threadIdx.x] += sh[threadIdx.x + off];
      sh2[threadIdx.x] += sh2[threadIdx.x + off];
    }
    __syncthreads();
  }
  if (threadIdx.x == 0) {
    part[(g * 64 + chunk) * 2] = sh[0];
    part[(g * 64 + chunk) * 2 + 1] = sh2[0];
  }
}

__global__ void gn_finalize_kernel(const float* __restrict__ part,
                                   float* __restrict__ stats, int Cout, int M) {
  int g = blockIdx.x;
  int tid = threadIdx.x;  // 64 threads
  __shared__ float sh[64], sh2[64];
  sh[tid] = part[(g * 64 + tid) * 2];
  sh2[tid] = part[(g * 64 + tid) * 2 + 1];
  __syncthreads();
  for (int off = 32; off > 0; off >>= 1) {
    if (tid < off) {
      sh[tid] += sh[tid + off];
      sh2[tid] += sh2[tid + off];
    }
    __syncthreads();
  }
  if (tid == 0) {
    size_t total = (size_t)(Cout / 4) * M;
    float inv = 1.f / (float)total;
    float mean = sh[0] * inv;
    float var = sh2[0] * inv - mean * mean;
    stats[g * 2] = mean;
    stats[g * 2 + 1] = rsqrtf(var + GN_EPS);
  }
}

__global__ void gn_lrelu_kernel(float* __restrict__ Y,
                                const float* __restrict__ stats,
                                const float* __restrict__ gamma,
                                const float* __restrict__ beta,
                                int Cout, int M) {
  size_t i = (size_t)blockIdx.x * blockDim.x + threadIdx.x;
  size_t total = (size_t)Cout * M;
  if (i >= total) return;
  int c = (int)(i / (size_t)M);
  int g = c / (Cout / 4);
  float v = (Y[i] - stats[g * 2]) * stats[g * 2 + 1] * gamma[c] + beta[c];
  Y[i] = v > 0.f ? v : LRELU_SLOPE * v;
}

// max over K=16 neighbors: out[c, q] = max_k Y[c, q*16+k]
__global__ void maxpool_kernel(const float* __restrict__ Y,
                               float* __restrict__ out, int C, int Nq) {
  int i = blockIdx.x * blockDim.x + threadIdx.x;
  if (i >= C * Nq) return;
  const float* p = Y + (size_t)i * 16;
  float m = p[0];
#pragma unroll
  for (int k = 1; k < 16; ++k) m = fmaxf(m, p[k]);
  out[i] = m;
}

// ---------------------------------------------------------------------------
extern "C" void kernel_launch(void* const* d_in, const int* in_sizes, int n_in,
                              void* d_out, int out_size, void* d_ws, size_t ws_size,
                              hipStream_t stream) {
  (void)n_in; (void)out_size; (void)ws_size;
  const int B = 8, N0 = 4096;

  const float* coor = (const float*)d_in[0];  // (B,3,4096)
  const float *w_in, *b_in, *w_out;
  const float *encW[3], *encG[3], *encB[3];
  const float *stW[2][3], *stG[2][3], *stB[2][3];
  if (in_sizes[2] == 48) {  // insertion-order flattening: w_in comes first
    w_in = (const float*)d_in[2];
    b_in = (const float*)d_in[3];
    int p = 4;
    for (int l = 0; l < 3; ++l) {
      encW[l] = (const float*)d_in[p++];
      encG[l] = (const float*)d_in[p++];
      encB[l] = (const float*)d_in[p++];
    }
    for (int s = 0; s < 2; ++s)
      for (int l = 0; l < 3; ++l) {
        stW[s][l] = (const float*)d_in[p++];
        stG[s][l] = (const float*)d_in[p++];
        stB[s][l] = (const float*)d_in[p++];
      }
    w_out = (const float*)d_in[p];
  } else {  // JAX pytree sorted keys: b_in, enc, stages, w_in, w_out
    b_in = (const float*)d_in[2];
    int p = 3;
    for (int l = 0; l < 3; ++l) {
      encW[l] = (const float*)d_in[p++];
      encG[l] = (const float*)d_in[p++];
      encB[l] = (const float*)d_in[p++];
    }
    for (int s = 0; s < 2; ++s)
      for (int l = 0; l < 3; ++l) {
        stW[s][l] = (const float*)d_in[p++];
        stG[s][l] = (const float*)d_in[p++];
        stB[s][l] = (const float*)d_in[p++];
      }
    w_in = (const float*)d_in[p++];
    w_out = (const float*)d_in[p];
  }

  // workspace carve (per-batch reuse; ~74 MB total)
  char* ws = (char*)d_ws;
  size_t off = 0;
  auto carve = [&](size_t bytes) -> void* {
    void* r = (void*)(ws + off);
    off = (off + bytes + 255) & ~(size_t)255;
    return r;
  };
  float* bufA   = (float*)carve(sizeof(float) * 128ull * 65536);  // conv out ping
  float* bufX   = (float*)carve(sizeof(float) * 132ull * 65536);  // X / pong
  float* feaA   = (float*)carve(sizeof(float) * 128 * 4096);
  float* feaB   = (float*)carve(sizeof(float) * 128 * 4096);
  int*   knnIdx = (int*)carve(sizeof(int) * 16 * 4096);
  int*   fpsIdx = (int*)carve(sizeof(int) * 4096);
  float* coorQ0 = (float*)carve(sizeof(float) * 3 * 1024);
  float* coorQ1 = (float*)carve(sizeof(float) * 3 * 256);
  float* stats  = (float*)carve(sizeof(float) * 8);
  float* part   = (float*)carve(sizeof(float) * 4 * 64 * 2);
  float* Wp     = (float*)carve(sizeof(float) * 256 * 132);

  auto conv_gn = [&](const float* W, const float* g, const float* bb,
                     int Cout, int Cin, int M, const float* Xp, float* Yp) {
    int Cin4 = (Cin + 3) & ~3;
    const float* Wuse = W;
    if (Cin4 != Cin) {
      pad_w_kernel<<<(Cout * Cin4 + 255) / 256, 256, 0, stream>>>(W, Wp, Cout,
                                                                  Cin, Cin4);
      Wuse = Wp;
    }
    conv1x1_wmma<<<dim3(M / 256, Cout / 16), dim3(128), 0, stream>>>(
        Wuse, Xp, Yp, Cout, Cin4, M);
    gn_partial_kernel<<<dim3(4, 64), dim3(256), 0, stream>>>(Yp, part, Cout, M);
    gn_finalize_kernel<<<dim3(4), dim3(64), 0, stream>>>(part, stats, Cout, M);
    size_t tot = (size_t)Cout * M;
    gn_lrelu_kernel<<<dim3((unsigned)((tot + 255) / 256)), dim3(256), 0, stream>>>(
        Yp, stats, g, bb, Cout, M);
  };

  float* out_cur = (float*)d_out;                   // (B,3,256)
  float* out_emb = out_cur + (size_t)B * 3 * 256;   // (B,256,256)

  for (int b = 0; b < B; ++b) {
    const float* cb = coor + (size_t)b * 3 * N0;

    // input embedding + self-kNN graph
    fea_in_kernel<<<(16 * N0 + 255) / 256, 256, 0, stream>>>(cb, w_in, b_in,
                                                             feaA, N0);
    knn_kernel<<<(N0 + 255) / 256, 256, 0, stream>>>(cb, N0, N0, cb, N0, N0,
                                                     knnIdx);
    int M = N0 * 16;
    graph_feat_kernel<<<(M + 255) / 256, 256, 0, stream>>>(
        cb, N0, N0, cb, N0, feaA, N0, 16, 20, knnIdx, bufX);

    // encoder MLP 19->32->64->128, then max over K
    conv_gn(encW[0], encG[0], encB[0], 32, 19, M, bufX, bufA);
    conv_gn(encW[1], encG[1], encB[1], 64, 32, M, bufA, bufX);
    conv_gn(encW[2], encG[2], encB[2], 128, 64, M, bufX, bufA);
    maxpool_kernel<<<(128 * N0 + 255) / 256, 256, 0, stream>>>(bufA, feaA, 128,
                                                               N0);

    const float* cur = cb;
    int curPitch = N0, Nk = N0;
    float* feaCur = feaA;
    float* feaNxt = feaB;
    float* cq[2] = {coorQ0, coorQ1};
    const int ms[2] = {1024, 256};

    for (int s = 0; s < 2; ++s) {
      int m = ms[s];
      fps_kernel<<<1, 256, 0, stream>>>(cur, curPitch, Nk, m, fpsIdx);
      gather_coor_kernel<<<(m + 255) / 256, 256, 0, stream>>>(cur, curPitch,
                                                              fpsIdx, cq[s], m);
      knn_kernel<<<(m + 255) / 256, 256, 0, stream>>>(cq[s], m, m, cur,
                                                      curPitch, Nk, knnIdx);
      int Ms = m * 16;
      graph_feat_kernel<<<(Ms + 255) / 256, 256, 0, stream>>>(
          cq[s], m, m, cur, curPitch, feaCur, Nk, 128, 132, knnIdx, bufX);

      conv_gn(stW[s][0], stG[s][0], stB[s][0], 128, 131, Ms, bufX, bufA);
      conv_gn(stW[s][1], stG[s][1], stB[s][1], 128, 128, Ms, bufA, bufX);
      conv_gn(stW[s][2], stG[s][2], stB[s][2], 128, 128, Ms, bufX, bufA);
      maxpool_kernel<<<(128 * m + 255) / 256, 256, 0, stream>>>(bufA, feaNxt,
                                                                128, m);
      cur = cq[s];
      curPitch = m;
      Nk = m;
      float* t = feaCur;
      feaCur = feaNxt;
      feaNxt = t;
    }

    // outputs: final coords + w_out projection (also WMMA, no GN)
    hipMemcpyAsync(out_cur + (size_t)b * 3 * 256, coorQ1,
                   sizeof(float) * 3 * 256, hipMemcpyDeviceToDevice, stream);
    conv1x1_wmma<<<dim3(1, 16), dim3(128), 0, stream>>>(
        w_out, feaCur, out_emb + (size_t)b * 256 * 256, 256, 128, 256);
  }
}